// PointMatcher_29437705847326
// MI455X (gfx1250) — compile-verified
//
#include <hip/hip_runtime.h>

typedef __attribute__((ext_vector_type(2))) float v2f;
typedef __attribute__((ext_vector_type(8))) float v8f;

#define N_PRED   1024
#define M_GT     2048
#define P        20
#define TILE     16
#define WAVES    8
#define JT_TOTAL (M_GT / TILE)        /* 128 gt tiles */
#define JT_PER_WAVE (JT_TOTAL / WAVES) /* 16 tiles per wave */

__global__ __launch_bounds__(256)
void PointMatcher_kernel(const float* __restrict__ pred,
                         const float* __restrict__ gt,
                         float* __restrict__ out_points,
                         float* __restrict__ out_conf,
                         int* __restrict__ out_idx)
{
    __shared__ float s_min[WAVES][TILE];
    __shared__ int   s_idx[WAVES][TILE];
    __shared__ int   s_final_idx[TILE];

    const int  tid    = threadIdx.x;
    const int  wave   = tid >> 5;
    const int  lane   = tid & 31;
    const int  col    = lane & 15;      // tile column (gt) / tile row (pred) for loads
    const bool hiHalf = lane >= 16;
    const int  ibase  = blockIdx.x * TILE;

    // ---- Build A fragments for all 20 points: (ax, ay | ||a||^2, 1) ----
    // 32-bit A 16x4 layout: lanes 0-15 = K0,K1 for M=lane; lanes 16-31 = K2,K3.
    v2f afrag[P];
#pragma unroll
    for (int p = 0; p < P; ++p) {
        const float* pp = pred + ((size_t)(ibase + col) * P + p) * 2;
        float ax = pp[0], ay = pp[1];
        if (!hiHalf) { afrag[p].x = ax;            afrag[p].y = ay;   }
        else         { afrag[p].x = ax*ax + ay*ay; afrag[p].y = 1.0f; }
    }

    float best[8];
    int   bidx[8];
#pragma unroll
    for (int r = 0; r < 8; ++r) { best[r] = 3.4e38f; bidx[r] = 0x7fffffff; }

    const v8f zero = {};

    for (int k = 0; k < JT_PER_WAVE; ++k) {
        const int jt    = wave * JT_PER_WAVE + k;
        const int jbase = jt * TILE;

        if (k + 1 < JT_PER_WAVE) {
            // global_prefetch_b8 of next gt tile (L2-resident anyway, but free)
            __builtin_prefetch(gt + ((size_t)(jbase + TILE + col) * P) * 2, 0, 3);
        }

        v8f sum = zero;
#pragma unroll
        for (int p = 0; p < P; ++p) {
            const float* gp = gt + ((size_t)(jbase + col) * P + p) * 2;
            float gx = gp[0], gy = gp[1];
            // B 4x16: lanes 0-15 = rows K0,K1 (-2bx,-2by); lanes 16-31 = K2,K3 (1, ||b||^2)
            v2f bfrag;
            if (!hiHalf) { bfrag.x = -2.0f * gx; bfrag.y = -2.0f * gy;   }
            else         { bfrag.x = 1.0f;       bfrag.y = gx*gx + gy*gy; }

            // D[i][j] = ||pred_i[p] - gt_j[p]||^2  for the 16x16 tile
            v8f d = __builtin_amdgcn_wmma_f32_16x16x4_f32(
                false, afrag[p], false, bfrag, (short)0, zero, false, false);

#pragma unroll
            for (int r = 0; r < 8; ++r) {
                float sq = fmaxf(d[r], 0.0f);   // guard tiny negative cancellation
                sum[r] += __builtin_amdgcn_sqrtf(sq);
            }
        }

        const int j = jbase + col;
#pragma unroll
        for (int r = 0; r < 8; ++r) {
            float mean = sum[r] * (1.0f / P);
            if (mean < best[r]) { best[r] = mean; bidx[r] = j; }
        }
    }

    // ---- Butterfly min/argmin across the 16 columns of each half-wave ----
#pragma unroll
    for (int mask = 1; mask < 16; mask <<= 1) {
#pragma unroll
        for (int r = 0; r < 8; ++r) {
            float ov = __shfl_xor(best[r], mask, 32);
            int   oi = __shfl_xor(bidx[r], mask, 32);
            if (ov < best[r] || (ov == best[r] && oi < bidx[r])) {
                best[r] = ov; bidx[r] = oi;
            }
        }
    }

    // lane 0 carries rows 0..7, lane 16 carries rows 8..15
    if (col == 0) {
#pragma unroll
        for (int r = 0; r < 8; ++r) {
            int row = r + (hiHalf ? 8 : 0);
            s_min[wave][row] = best[r];
            s_idx[wave][row] = bidx[r];
        }
    }
    __syncthreads();

    // ---- Cross-wave reduction; one thread per output row ----
    if (tid < TILE) {
        float m  = s_min[0][tid];
        int   ix = s_idx[0][tid];
        for (int w = 1; w < WAVES; ++w) {
            float ov = s_min[w][tid];
            int   oi = s_idx[w][tid];
            if (ov < m || (ov == m && oi < ix)) { m = ov; ix = oi; }
        }
        s_final_idx[tid] = ix;

        float conf = (m > 2.0f) ? 0.0f
                   : __builtin_amdgcn_exp2f(-m * 1.4426950408889634f);
        out_conf[ibase + tid] = conf;
        out_idx [ibase + tid] = ix;
    }
    __syncthreads();

    // ---- Gather matched gt points: 16 rows x 40 floats ----
    for (int e = tid; e < TILE * P * 2; e += 256) {
        int row = e / (P * 2);
        int off = e % (P * 2);
        int gi  = s_final_idx[row];
        out_points[(size_t)(ibase + row) * (P * 2) + off] =
            gt[(size_t)gi * (P * 2) + off];
    }
}

extern "C" void kernel_launch(void* const* d_in, const int* in_sizes, int n_in,
                              void* d_out, int out_size, void* d_ws, size_t ws_size,
                              hipStream_t stream) {
    const float* pred = (const float*)d_in[0];   // (1024, 20, 2) f32
    const float* gt   = (const float*)d_in[1];   // (2048, 20, 2) f32
    float* out        = (float*)d_out;
    float* out_points = out;                         // 1024*20*2 = 40960 f32
    float* out_conf   = out + (size_t)N_PRED * P * 2; // 1024 f32
    int*   out_idx    = (int*)(out + (size_t)N_PRED * P * 2 + N_PRED); // 1024 i32

    PointMatcher_kernel<<<N_PRED / TILE, 256, 0, stream>>>(
        pred, gt, out_points, out_conf, out_idx);
}